// GatingFunc_85590108275211
// MI455X (gfx1250) — compile-verified
//
#include <hip/hip_runtime.h>

typedef __attribute__((ext_vector_type(2))) float v2f;
typedef __attribute__((ext_vector_type(4))) float v4f;
typedef __attribute__((ext_vector_type(8))) float v8f;

#define D_MODEL      1024
#define NUM_EXPERTS  64
#define N_TOKENS     131072
#define WROW         1028   // padded LDS row stride (dwords) for W: 1028 % 64 == 4 -> conflict-free b64 B loads
#define LROW         68     // padded LDS row stride (dwords) for logits tile
#define TOK_PER_BLK  128
#define THREADS      256

__global__ __launch_bounds__(THREADS, 1)
void moe_gating_wmma_kernel(const float* __restrict__ x,
                            const float* __restrict__ W,
                            const float* __restrict__ b,
                            float* __restrict__ out) {
  extern __shared__ float smem[];
  float* wlds = smem;                                  // 64 * 1028 floats  (W, padded rows)
  float* llds = smem + NUM_EXPERTS * WROW;             // 128 * 68 floats   (logits tile)
  float* t2   = llds + TOK_PER_BLK * LROW;             // 128 * 4 floats    (i1, i2, w0, w1)

  const int tid  = threadIdx.x;
  const int lane = tid & 31;
  const int wave = tid >> 5;       // 8 wave32s
  const int half = lane >> 4;      // 0: K pair 0..1 / M 0..7 ; 1: K pair 2..3 / M 8..15
  const int quad = lane & 15;      // M (for A) or N (for B/C/D) within the 16x16 tile
  const int blockTok = blockIdx.x * TOK_PER_BLK;

  // ---- Stage W [64 x 1024] row-major into padded LDS (coalesced float4 loads) ----
  {
    const v4f* Wv = (const v4f*)W;
    for (int f = tid; f < NUM_EXPERTS * D_MODEL / 4; f += THREADS) {
      int e   = f >> 8;                     // (f*4) / 1024
      int col = (f << 2) & (D_MODEL - 1);   // (f*4) % 1024
      *(v4f*)(wlds + e * WROW + col) = Wv[f];
    }
  }
  __syncthreads();

  // ---- f32 WMMA matmul: this wave computes logits for tokens [tb, tb+16) x all 64 experts ----
  const int tb = blockTok + wave * 16;
  const float* xrow = x + (size_t)(tb + quad) * D_MODEL + 2 * half;

  const float* bl0 = wlds + ( 0 + quad) * WROW + 2 * half;
  const float* bl1 = wlds + (16 + quad) * WROW + 2 * half;
  const float* bl2 = wlds + (32 + quad) * WROW + 2 * half;
  const float* bl3 = wlds + (48 + quad) * WROW + 2 * half;

  v8f acc0 = {}, acc1 = {}, acc2 = {}, acc3 = {};

  #pragma unroll 4
  for (int kk = 0; kk < D_MODEL; kk += 4) {
    v2f a  = *(const v2f*)(xrow + kk);   // A 16x4 f32 fragment (ISA layout)
    v2f b0 = *(const v2f*)(bl0 + kk);    // B 4x16 f32 fragments from LDS
    v2f b1 = *(const v2f*)(bl1 + kk);
    v2f b2 = *(const v2f*)(bl2 + kk);
    v2f b3 = *(const v2f*)(bl3 + kk);
    acc0 = __builtin_amdgcn_wmma_f32_16x16x4_f32(false, a, false, b0, (short)0, acc0, false, false);
    acc1 = __builtin_amdgcn_wmma_f32_16x16x4_f32(false, a, false, b1, (short)0, acc1, false, false);
    acc2 = __builtin_amdgcn_wmma_f32_16x16x4_f32(false, a, false, b2, (short)0, acc2, false, false);
    acc3 = __builtin_amdgcn_wmma_f32_16x16x4_f32(false, a, false, b3, (short)0, acc3, false, false);
  }

  // ---- Add bias (expert index = 16*t + quad, same for all 8 rows of a lane) ----
  acc0 += b[ 0 + quad];
  acc1 += b[16 + quad];
  acc2 += b[32 + quad];
  acc3 += b[48 + quad];

  // ---- Spill logits tile to LDS (C/D layout: VGPR j holds M = 8*half + j, N = quad) ----
  #pragma unroll
  for (int j = 0; j < 8; ++j) {
    int m = wave * 16 + half * 8 + j;
    float* p = llds + m * LROW + quad;
    p[ 0] = acc0[j];
    p[16] = acc1[j];
    p[32] = acc2[j];
    p[48] = acc3[j];
  }
  __syncthreads();

  // ---- Top-2 + softmax: one thread per token (stable tie-break like jax top_k) ----
  if (tid < TOK_PER_BLK) {
    const float* row = llds + tid * LROW;
    float v1 = -3.402823466e38f, v2s = -3.402823466e38f;
    int i1 = 0, i2 = 0;
    #pragma unroll
    for (int c4 = 0; c4 < NUM_EXPERTS; c4 += 4) {
      v4f vv = *(const v4f*)(row + c4);
      #pragma unroll
      for (int j = 0; j < 4; ++j) {
        float v = vv[j];
        int   c = c4 + j;
        if (v > v1)       { v2s = v1; i2 = i1; v1 = v; i1 = c; }
        else if (v > v2s) { v2s = v;  i2 = c; }
      }
    }
    float w0 = 1.0f / (1.0f + __expf(v2s - v1));  // softmax over [v1, v2s], v2s <= v1
    t2[tid * 4 + 0] = (float)i1;
    t2[tid * 4 + 1] = (float)i2;
    t2[tid * 4 + 2] = w0;
    t2[tid * 4 + 3] = 1.0f - w0;
  }
  __syncthreads();

  // ---- Scatter to dense sparse output, coalesced float4 stores (128 x 64 tile) ----
  for (int q = tid; q < TOK_PER_BLK * (NUM_EXPERTS / 4); q += THREADS) {
    int r  = q >> 4;              // token row in block
    int c4 = (q & 15) << 2;       // starting expert column
    int   i1 = (int)t2[r * 4 + 0];
    int   i2 = (int)t2[r * 4 + 1];
    float w0 =      t2[r * 4 + 2];
    float w1 =      t2[r * 4 + 3];
    v4f o;
    #pragma unroll
    for (int j = 0; j < 4; ++j) {
      int c = c4 + j;
      o[j] = (c == i1) ? w0 : ((c == i2) ? w1 : 0.0f);
    }
    *(v4f*)(out + (size_t)(blockTok + r) * NUM_EXPERTS + c4) = o;
  }
}

extern "C" void kernel_launch(void* const* d_in, const int* in_sizes, int n_in,
                              void* d_out, int out_size, void* d_ws, size_t ws_size,
                              hipStream_t stream) {
  const float* x = (const float*)d_in[0];
  const float* W = (const float*)d_in[1];
  const float* b = (const float*)d_in[2];
  float* out = (float*)d_out;

  size_t shmem = (size_t)(NUM_EXPERTS * WROW + TOK_PER_BLK * LROW + TOK_PER_BLK * 4) * sizeof(float);
  dim3 grid(N_TOKENS / TOK_PER_BLK);   // 1024 workgroups x 128 tokens
  moe_gating_wmma_kernel<<<grid, THREADS, shmem, stream>>>(x, W, b, out);
}